// LossFunc_40346922779395
// MI455X (gfx1250) — compile-verified
//
#include <hip/hip_runtime.h>
#include <cstdint>

// Problem constants (from reference)
#define B_DIM 64
#define T_DIM 188
#define V_DIM 32000
#define ROWS (B_DIM * T_DIM)   // 12032
#define NV4 (V_DIM / 4)        // 8000 float4 per row
#define NSTEPS (NV4 / 32)      // 250 steps of 32 float4 (512B) per row
#define WAVES 8                // 256 threads
#define NBUF 6                 // LDS ring buffers per wave
#define INFLIGHT 4             // async ops in flight per wave
#define BETA 2.0f

// ---- CDNA5 async global->LDS helpers (gfx1250) ------------------------------
__device__ __forceinline__ void async_load_b128(uint32_t lds_off, const void* gaddr) {
    // GLOBAL_LOAD_ASYNC_TO_LDS_B128 : per-lane copy of 16B from the global address in vaddr to the LDS address in vdst
    asm volatile("global_load_async_to_lds_b128 %0, %1, off"
                 :: "v"(lds_off), "v"(gaddr)
                 : "memory");
}
#define WAIT_ASYNCCNT(n) asm volatile("s_wait_asynccnt " #n ::: "memory")
#define WAIT_DSCNT0()    asm volatile("s_wait_dscnt 0" ::: "memory")

// ---- Kernel 1: per-row argmax + gathered-log partial loss -------------------
__global__ __launch_bounds__(256) void row_loss_kernel(
    const float* __restrict__ scores,   // [B,T,V]
    const int*   __restrict__ tin,      // [B,T]
    const int*   __restrict__ lens,     // [B]
    float*       __restrict__ partials) // [B*T]
{
    const int row  = blockIdx.x;
    const int tid  = threadIdx.x;
    const int lane = tid & 31;
    const int w    = tid >> 5;
    const float* rowp = scores + (size_t)row * V_DIM;

    // 8 waves x 6 bufs x 32 lanes x 16B = 24 KB
    __shared__ float4 sbuf[WAVES][NBUF][32];
    __shared__ float  swv[WAVES];
    __shared__ int    swi[WAVES];

    // Wave w owns global steps s = w, w+8, ... ; wave-local step k -> s = w + 8k
    const int myN = (NSTEPS - w + WAVES - 1) / WAVES;   // 32 or 31

    // element index for wave-local step k, this lane: eBase + k*1024
    const int eBase = (w * 32 + lane) * 4;
    const float* g0 = rowp + eBase;                     // k=0 address (16B aligned)

    float bestVal  = -1.0f;                             // scores are > 0
    int   bestBase = 0;                                 // element idx of winning quad

    // ---- prologue: fill pipeline with INFLIGHT steps ----
    const int nIss = (myN < INFLIGHT) ? myN : INFLIGHT;
    for (int p = 0; p < nIss; ++p) {
        const uint32_t ldso = (uint32_t)(uintptr_t)&sbuf[w][p][lane];
        async_load_b128(ldso, g0 + (size_t)p * 1024);
    }

    // ---- steady state: 2 steps per iteration, issue decoupled from consume ----
    int t  = 0;
    int qi = nIss % NBUF;   // next issue buffer
    int ci = 0;             // next consume buffer
    for (; t + 5 < myN; t += 2) {
        WAIT_DSCNT0();      // prev iteration's ds_loads done -> bufs reusable
        {
            const uint32_t l0 = (uint32_t)(uintptr_t)&sbuf[w][qi][lane];
            async_load_b128(l0, g0 + (size_t)(t + 4) * 1024);
            int q2 = qi + 1; if (q2 == NBUF) q2 = 0;
            const uint32_t l1 = (uint32_t)(uintptr_t)&sbuf[w][q2][lane];
            async_load_b128(l1, g0 + (size_t)(t + 5) * 1024);
            qi = q2 + 1; if (qi == NBUF) qi = 0;
        }
        WAIT_ASYNCCNT(4);   // issued = t+6; ensures steps t, t+1 have landed
        {
            const float4 v0 = sbuf[w][ci][lane];
            int c2 = ci + 1; if (c2 == NBUF) c2 = 0;
            const float4 v1 = sbuf[w][c2][lane];
            ci = c2 + 1; if (ci == NBUF) ci = 0;

            const float m0 = fmaxf(fmaxf(v0.x, v0.y), fmaxf(v0.z, v0.w));
            if (m0 > bestVal) { bestVal = m0; bestBase = eBase + t * 1024; }
            const float m1 = fmaxf(fmaxf(v1.x, v1.y), fmaxf(v1.z, v1.w));
            if (m1 > bestVal) { bestVal = m1; bestBase = eBase + (t + 1) * 1024; }
        }
    }

    // ---- epilogue: issue the (at most one) remaining step, then drain ----
    WAIT_DSCNT0();
    for (int s = t + INFLIGHT; s < myN; ++s) {          // issued total was t+4
        const uint32_t l0 = (uint32_t)(uintptr_t)&sbuf[w][qi][lane];
        async_load_b128(l0, g0 + (size_t)s * 1024);
        if (++qi == NBUF) qi = 0;
    }
    WAIT_ASYNCCNT(0);
    for (; t < myN; ++t) {
        const float4 v = sbuf[w][ci][lane];
        if (++ci == NBUF) ci = 0;
        const float m = fmaxf(fmaxf(v.x, v.y), fmaxf(v.z, v.w));
        if (m > bestVal) { bestVal = m; bestBase = eBase + t * 1024; }
    }

    // ---- resolve first-occurrence index inside winning quad (global re-read) ----
    const float4 bv = *(const float4*)(rowp + bestBase);   // bit-identical data
    int bestIdx;
    if      (bv.x == bestVal) bestIdx = bestBase;
    else if (bv.y == bestVal) bestIdx = bestBase + 1;
    else if (bv.z == bestVal) bestIdx = bestBase + 2;
    else                      bestIdx = bestBase + 3;

    // ---- wave32 butterfly reduce (max value, min index on ties) ----
    #pragma unroll
    for (int off = 16; off; off >>= 1) {
        const float ov = __shfl_xor(bestVal, off, 32);
        const int   oi = __shfl_xor(bestIdx, off, 32);
        if (ov > bestVal || (ov == bestVal && oi < bestIdx)) { bestVal = ov; bestIdx = oi; }
    }
    if (lane == 0) { swv[w] = bestVal; swi[w] = bestIdx; }
    __syncthreads();

    if (tid == 0) {
        float bV = swv[0]; int bI = swi[0];
        #pragma unroll
        for (int i = 1; i < WAVES; ++i) {
            const float vv = swv[i]; const int ii = swi[i];
            if (vv > bV || (vv == bV && ii < bI)) { bV = vv; bI = ii; }
        }
        const int target = tin[row];
        const int b  = row / T_DIM;
        const int tt = row - b * T_DIM;
        const int len = lens[b];
        const float g = rowp[target];                     // row hot in L2
        const float wgt = (bI == target && target != 0) ? BETA : 1.0f;
        partials[row] = (tt < len) ? -(wgt * logf(g)) : 0.0f;
    }
}

// ---- Kernel 2: deterministic fixed-order final reduction --------------------
__global__ __launch_bounds__(256) void final_reduce_kernel(
    const float* __restrict__ partials, int n, float* __restrict__ out)
{
    __shared__ float s[WAVES];
    const int tid  = threadIdx.x;
    const int lane = tid & 31;
    const int w    = tid >> 5;

    float acc = 0.0f;
    for (int i = tid; i < n; i += 256) acc += partials[i];

    #pragma unroll
    for (int off = 16; off; off >>= 1) acc += __shfl_xor(acc, off, 32);
    if (lane == 0) s[w] = acc;
    __syncthreads();

    if (tid == 0) {
        float t = 0.0f;
        #pragma unroll
        for (int i = 0; i < WAVES; ++i) t += s[i];
        out[0] = t / (float)B_DIM;
    }
}

// ---- Host entry -------------------------------------------------------------
extern "C" void kernel_launch(void* const* d_in, const int* in_sizes, int n_in,
                              void* d_out, int out_size, void* d_ws, size_t ws_size,
                              hipStream_t stream) {
    (void)in_sizes; (void)n_in; (void)out_size; (void)ws_size;
    const float* scores = (const float*)d_in[0];
    const int*   tin    = (const int*)d_in[1];
    const int*   lens   = (const int*)d_in[2];
    float* partials = (float*)d_ws;        // ROWS floats, fully rewritten each launch
    float* out      = (float*)d_out;

    row_loss_kernel<<<ROWS, 256, 0, stream>>>(scores, tin, lens, partials);
    final_reduce_kernel<<<1, 256, 0, stream>>>(partials, ROWS, out);
}